// SelfAttention_61314953118402
// MI455X (gfx1250) — compile-verified
//
#include <hip/hip_runtime.h>
#include <math.h>

// ---------------------------------------------------------------------------
// Performer-style linear attention, MI455X (gfx1250), wave32, f32 WMMA path.
// Bandwidth-bound (~1 GB traffic / 23.3 TB/s); V_WMMA_F32_16X16X4_F32 keeps
// reference f32 precision while putting all GEMMs on the matrix pipe.
// This revision: vectorized (b64/b128) global<->LDS staging, dual WMMA
// accumulator chains, float4 split-K reduce, tile prefetch.
// ---------------------------------------------------------------------------

typedef __attribute__((ext_vector_type(2))) float v2f;
typedef __attribute__((ext_vector_type(8))) float v8f;

#define BATCH 4
#define SEQ   4096
#define DMODEL 512
#define HEADS 8
#define DHEAD 64
#define MPROJ 128          // kernel-map size; features = 2*MPROJ = 256
#define ROWS  (BATCH*SEQ)  // 16384
#define RH    (ROWS*HEADS) // 131072
#define KV_SPLIT 16

__device__ __forceinline__ v8f wmma4(v2f a, v2f b, v8f c) {
    // D = A(16x4,f32) * B(4x16,f32) + C(16x16,f32)
    return __builtin_amdgcn_wmma_f32_16x16x4_f32(
        /*neg_a=*/false, a, /*neg_b=*/false, b,
        /*c_mod=*/(short)0, c, /*reuse_a=*/false, /*reuse_b=*/false);
}

// ---------------------------------------------------------------------------
// Generic GEMM: C[r][n] = sum_k A[r][k] * W[k][n], K = N = 512.
// Block: 256 threads = 8 waves; block tile 32x64; wave tile 16x16.
// grid = (ROWS/32, 512/64)
// ---------------------------------------------------------------------------
__global__ __launch_bounds__(256) void gemm_rk512(
    const float* __restrict__ A, const float* __restrict__ W,
    float* __restrict__ C) {
    __shared__ float As[32][18];   // pitch 18: b64-aligned, conflict-light
    __shared__ float Bs[16][72];   // pitch 72: b128-aligned, halves disjoint
    const int tid  = threadIdx.x;
    const int wave = tid >> 5, lane = tid & 31;
    const int wr = wave >> 2, wc = wave & 3;
    const int half = lane >> 4, n = lane & 15;
    const int row0 = blockIdx.x * 32;
    const int col0 = blockIdx.y * 64;

    const int ar = tid >> 3, ak = (tid & 7) * 2;     // A: 2 contiguous floats
    const int bk = tid >> 4, bc = (tid & 15) * 4;    // B: 4 contiguous floats

    v8f acc0 = {}, acc1 = {};
    for (int k0 = 0; k0 < 512; k0 += 16) {
        __syncthreads();
        *reinterpret_cast<float2*>(&As[ar][ak]) =
            *reinterpret_cast<const float2*>(&A[(size_t)(row0 + ar) * 512 + k0 + ak]);
        *reinterpret_cast<float4*>(&Bs[bk][bc]) =
            *reinterpret_cast<const float4*>(&W[(size_t)(k0 + bk) * 512 + col0 + bc]);
        if (k0 + 16 < 512) {   // prefetch next tiles
            __builtin_prefetch(&A[(size_t)(row0 + ar) * 512 + k0 + 16 + ak], 0, 1);
            __builtin_prefetch(&W[(size_t)(k0 + 16 + bk) * 512 + col0 + bc], 0, 1);
        }
        __syncthreads();
        #pragma unroll
        for (int kk = 0; kk < 8; kk += 4) {          // two independent chains
            v2f a, b;
            a[0] = As[wr * 16 + n][kk + 2 * half + 0];
            a[1] = As[wr * 16 + n][kk + 2 * half + 1];
            b[0] = Bs[kk + 2 * half + 0][wc * 16 + n];
            b[1] = Bs[kk + 2 * half + 1][wc * 16 + n];
            acc0 = wmma4(a, b, acc0);
            a[0] = As[wr * 16 + n][kk + 8 + 2 * half + 0];
            a[1] = As[wr * 16 + n][kk + 8 + 2 * half + 1];
            b[0] = Bs[kk + 8 + 2 * half + 0][wc * 16 + n];
            b[1] = Bs[kk + 8 + 2 * half + 1][wc * 16 + n];
            acc1 = wmma4(a, b, acc1);
        }
    }
    const int crow = row0 + wr * 16 + 8 * half;
    const int ccol = col0 + wc * 16 + n;
    #pragma unroll
    for (int i = 0; i < 8; ++i)
        C[(size_t)(crow + i) * 512 + ccol] = acc0[i] + acc1[i];
}

// ---------------------------------------------------------------------------
// Row norms of slopes_proj: norms[rho] = ||slopes_proj[rho,:64]|| / SEQ
// One wave per row. grid = RH/8, block = 256.
// ---------------------------------------------------------------------------
__global__ __launch_bounds__(256) void norms_kernel(
    const float* __restrict__ sp, float* __restrict__ norms) {
    const int row  = blockIdx.x * 8 + (threadIdx.x >> 5);
    const int lane = threadIdx.x & 31;
    const float2 p = *reinterpret_cast<const float2*>(&sp[(size_t)row * 64 + lane * 2]);
    float s = p.x * p.x + p.y * p.y;
    #pragma unroll
    for (int off = 16; off > 0; off >>= 1) s += __shfl_xor(s, off, 32);
    if (lane == 0) norms[row] = sqrtf(s) / (float)SEQ;
}

// ---------------------------------------------------------------------------
// Fourier features: dash = (nrm * data) @ proj^T ; out = ratio*[sin,cos].
// Rows are rho = (b*SEQ+l)*HEADS + h (RH total, length 64), cols = 128.
// Block: 16 rows x 128 cols (8 waves, one 16x16 tile each), computes q AND k
// tiles sharing the B fragments. Writes directly into the transposed
// [L,B,H,2M] output layout. grid = RH/16.
// ---------------------------------------------------------------------------
__global__ __launch_bounds__(256) void fourier_kernel(
    const float* __restrict__ pos_proj, const float* __restrict__ slopes_proj,
    const float* __restrict__ scale, const float* __restrict__ offsets,
    const float* __restrict__ proj,
    float* __restrict__ qprime, float* __restrict__ kprime) {
    __shared__ float Bp[64][130];   // proj^T : Bp[k][m]
    __shared__ float qA[16][68];
    __shared__ float kA[16][68];

    const float nrm   = 0.35355339059327378f;   // 64^-0.25
    const float ratio = 0.08838834764831845f;   // 1/sqrt(128)

    const int tid  = threadIdx.x;
    const int wave = tid >> 5, lane = tid & 31;
    const int half = lane >> 4, n = lane & 15;
    const int rho0 = blockIdx.x * 16;

    #pragma unroll
    for (int it = 0; it < 8; ++it) {            // proj^T into LDS, b128 reads
        int e = it * 1024 + tid * 4;
        int m = e >> 6, k = e & 63;
        float4 p4 = *reinterpret_cast<const float4*>(&proj[m * 64 + k]);
        Bp[k + 0][m] = p4.x; Bp[k + 1][m] = p4.y;
        Bp[k + 2][m] = p4.z; Bp[k + 3][m] = p4.w;
    }
    {                                           // load & pre-scale A rows
        int e = tid * 4;                        // 1024 elems, contiguous
        int i = e >> 6, k = e & 63;
        int h = (rho0 + i) & 7;
        float sc = scale[h], of = offsets[h];
        float4 pv = *reinterpret_cast<const float4*>(&pos_proj[(size_t)rho0 * 64 + e]);
        float4 sv = *reinterpret_cast<const float4*>(&slopes_proj[(size_t)rho0 * 64 + e]);
        qA[i][k + 0] = nrm * sc * pv.x;
        qA[i][k + 1] = nrm * sc * pv.y;
        qA[i][k + 2] = nrm * sc * pv.z;
        qA[i][k + 3] = nrm * sc * pv.w;
        kA[i][k + 0] = nrm * sc * (pv.x + of * sv.x);
        kA[i][k + 1] = nrm * sc * (pv.y + of * sv.y);
        kA[i][k + 2] = nrm * sc * (pv.z + of * sv.z);
        kA[i][k + 3] = nrm * sc * (pv.w + of * sv.w);
    }
    __syncthreads();

    v8f accq = {}, acck = {};
    #pragma unroll
    for (int kk = 0; kk < 64; kk += 4) {
        v2f aq, ak, b;
        aq[0] = qA[n][kk + 2 * half + 0];
        aq[1] = qA[n][kk + 2 * half + 1];
        ak[0] = kA[n][kk + 2 * half + 0];
        ak[1] = kA[n][kk + 2 * half + 1];
        b[0]  = Bp[kk + 2 * half + 0][wave * 16 + n];
        b[1]  = Bp[kk + 2 * half + 1][wave * 16 + n];
        accq = wmma4(aq, b, accq);
        acck = wmma4(ak, b, acck);
    }

    const int m = wave * 16 + n;                // global feature column 0..127
    #pragma unroll
    for (int i = 0; i < 8; ++i) {
        int rho = rho0 + i + 8 * half;
        int h = rho & 7, t = rho >> 3;
        int l = t & (SEQ - 1), b = t >> 12;
        size_t base = ((size_t)(l * BATCH + b) * HEADS + h) * (2 * MPROJ);
        float s, c;
        __sincosf(accq[i], &s, &c);
        qprime[base + m]         = ratio * s;
        qprime[base + MPROJ + m] = ratio * c;
        __sincosf(acck[i], &s, &c);
        kprime[base + m]         = ratio * s;
        kprime[base + MPROJ + m] = ratio * c;
    }
}

// ---------------------------------------------------------------------------
// kvs partials: for (b,h), split s: part[s,bh,m,d] = sum_{l in chunk}
// k'[b,l,h,m] * value[b,l,h,d].   A = k'^T (256 x L), B = V (L x 64).
// grid = (32, KV_SPLIT); block 256 = 8 waves; each wave: 2 m-tiles x 4 d-tiles
// ---------------------------------------------------------------------------
__global__ __launch_bounds__(256) void kvs_partial_kernel(
    const float* __restrict__ kprime, const float* __restrict__ value,
    float* __restrict__ part) {
    __shared__ float Kt[16][264];   // Kt[l][m]; pitch 264: b128 + bank-clean
    __shared__ float Vt[16][72];    // Vt[l][d]

    const int tid  = threadIdx.x;
    const int wave = tid >> 5, lane = tid & 31;
    const int half = lane >> 4, n = lane & 15;
    const int bh = blockIdx.x;
    const int b = bh >> 3, h = bh & 7;
    const int l0 = blockIdx.y * (SEQ / KV_SPLIT);

    v8f acc[2][4];
    #pragma unroll
    for (int mt = 0; mt < 2; ++mt)
        #pragma unroll
        for (int nt = 0; nt < 4; ++nt) acc[mt][nt] = (v8f){};

    const int vli = tid >> 4, vd = (tid & 15) * 4;   // V: 4 contiguous floats

    for (int kc = 0; kc < SEQ / KV_SPLIT; kc += 16) {
        __syncthreads();
        #pragma unroll
        for (int g = 0; g < 4; ++g) {            // K': 4x b128 per thread
            int e = g * 1024 + tid * 4;
            int li = e >> 8, m = e & 255;
            int l = l0 + kc + li;
            *reinterpret_cast<float4*>(&Kt[li][m]) =
                *reinterpret_cast<const float4*>(
                    &kprime[((size_t)(l * BATCH + b) * HEADS + h) * 256 + m]);
        }
        {
            int l = l0 + kc + vli;
            *reinterpret_cast<float4*>(&Vt[vli][vd]) =
                *reinterpret_cast<const float4*>(
                    &value[((size_t)(b * SEQ + l) * HEADS + h) * 64 + vd]);
        }
        if (kc + 16 < SEQ / KV_SPLIT) {
            int lp = l0 + kc + 16 + vli;
            __builtin_prefetch(&value[((size_t)(b * SEQ + lp) * HEADS + h) * 64 + vd], 0, 1);
        }
        __syncthreads();
        #pragma unroll
        for (int kk = 0; kk < 16; kk += 4) {
            #pragma unroll
            for (int mt = 0; mt < 2; ++mt) {
                v2f a;
                a[0] = Kt[kk + 2 * half + 0][(wave * 2 + mt) * 16 + n];
                a[1] = Kt[kk + 2 * half + 1][(wave * 2 + mt) * 16 + n];
                #pragma unroll
                for (int nt = 0; nt < 4; ++nt) {
                    v2f bb;
                    bb[0] = Vt[kk + 2 * half + 0][nt * 16 + n];
                    bb[1] = Vt[kk + 2 * half + 1][nt * 16 + n];
                    acc[mt][nt] = wmma4(a, bb, acc[mt][nt]);
                }
            }
        }
    }
    float* dst = part + ((size_t)blockIdx.y * 32 + bh) * (256 * 64);
    #pragma unroll
    for (int mt = 0; mt < 2; ++mt)
        #pragma unroll
        for (int nt = 0; nt < 4; ++nt)
            #pragma unroll
            for (int i = 0; i < 8; ++i) {
                int m = (wave * 2 + mt) * 16 + i + 8 * half;
                int d = nt * 16 + n;
                dst[m * 64 + d] = acc[mt][nt][i];
            }
}

// Deterministic split-K reduce: kvs[j] = sum_s part[s][j]   (float4-wide)
__global__ __launch_bounds__(256) void kvs_reduce_kernel(
    const float4* __restrict__ part, float4* __restrict__ kvs) {
    size_t j = (size_t)blockIdx.x * 256 + threadIdx.x;   // 131072 float4s
    float4 s = make_float4(0.f, 0.f, 0.f, 0.f);
    #pragma unroll
    for (int sp = 0; sp < KV_SPLIT; ++sp) {
        float4 p = part[(size_t)sp * 131072 + j];
        s.x += p.x; s.y += p.y; s.z += p.z; s.w += p.w;
    }
    kvs[j] = s;
}

// ---------------------------------------------------------------------------
// av[b,l,h,d] = norms[b,l,h] * sum_m q'[b,l,h,m] * kvs[b,h,m,d]
// grid = (SEQ/32, 32); block tile 32 l-rows x 64 d-cols; K = 256.
// av stored as [ROWS][512] row-major (rho*64+d) so the output GEMM reuses it.
// ---------------------------------------------------------------------------
__global__ __launch_bounds__(256) void av_kernel(
    const float* __restrict__ qprime, const float* __restrict__ kvs,
    const float* __restrict__ norms, float* __restrict__ av) {
    __shared__ float As[32][18];
    __shared__ float Bs[16][72];
    const int tid  = threadIdx.x;
    const int wave = tid >> 5, lane = tid & 31;
    const int wr = wave >> 2, wc = wave & 3;
    const int half = lane >> 4, n = lane & 15;
    const int l0 = blockIdx.x * 32;
    const int bh = blockIdx.y;
    const int b = bh >> 3, h = bh & 7;

    const int ar = tid >> 3, ak = (tid & 7) * 2;
    const int bk = tid >> 4, bc = (tid & 15) * 4;

    v8f acc0 = {}, acc1 = {};
    for (int k0 = 0; k0 < 256; k0 += 16) {
        __syncthreads();
        {
            int l = l0 + ar;
            *reinterpret_cast<float2*>(&As[ar][ak]) =
                *reinterpret_cast<const float2*>(
                    &qprime[((size_t)(l * BATCH + b) * HEADS + h) * 256 + k0 + ak]);
        }
        *reinterpret_cast<float4*>(&Bs[bk][bc]) =
            *reinterpret_cast<const float4*>(
                &kvs[(size_t)bh * (256 * 64) + (k0 + bk) * 64 + bc]);
        __syncthreads();
        #pragma unroll
        for (int kk = 0; kk < 8; kk += 4) {
            v2f a, bb;
            a[0]  = As[wr * 16 + n][kk + 2 * half + 0];
            a[1]  = As[wr * 16 + n][kk + 2 * half + 1];
            bb[0] = Bs[kk + 2 * half + 0][wc * 16 + n];
            bb[1] = Bs[kk + 2 * half + 1][wc * 16 + n];
            acc0 = wmma4(a, bb, acc0);
            a[0]  = As[wr * 16 + n][kk + 8 + 2 * half + 0];
            a[1]  = As[wr * 16 + n][kk + 8 + 2 * half + 1];
            bb[0] = Bs[kk + 8 + 2 * half + 0][wc * 16 + n];
            bb[1] = Bs[kk + 8 + 2 * half + 1][wc * 16 + n];
            acc1 = wmma4(a, bb, acc1);
        }
    }
    #pragma unroll
    for (int i = 0; i < 8; ++i) {
        int l = l0 + wr * 16 + i + 8 * half;
        size_t rho = (size_t)(b * SEQ + l) * HEADS + h;
        av[rho * 64 + wc * 16 + n] = norms[rho] * (acc0[i] + acc1[i]);
    }
}

// ---------------------------------------------------------------------------
extern "C" void kernel_launch(void* const* d_in, const int* in_sizes, int n_in,
                              void* d_out, int out_size, void* d_ws, size_t ws_size,
                              hipStream_t stream) {
    (void)in_sizes; (void)n_in; (void)out_size; (void)ws_size;

    const float* src    = (const float*)d_in[0];   // [B,L,D]
    const float* pos    = (const float*)d_in[1];   // [B,L,D]
    const float* slo    = (const float*)d_in[2];   // [B,L,D]
    const float* Wv     = (const float*)d_in[3];   // [D,H,DH] = [512,512]
    const float* Wp     = (const float*)d_in[4];   // [D,H,DH]
    const float* scale  = (const float*)d_in[5];   // [H]
    const float* offs   = (const float*)d_in[6];   // [H]
    const float* Wo     = (const float*)d_in[7];   // [512,512]
    const float* proj   = (const float*)d_in[8];   // [M,DH] = [128,64]

    float* out    = (float*)d_out;                       // [B,L,D]
    float* qprime = out + (size_t)ROWS * DMODEL;         // [L,B,H,2M]
    float* kprime = qprime + (size_t)SEQ * BATCH * HEADS * 2 * MPROJ;

    float* ws          = (float*)d_ws;
    float* value       = ws;                                  // ROWS*512
    float* pos_proj    = value + (size_t)ROWS * 512;          // ROWS*512
    float* slopes_proj = pos_proj + (size_t)ROWS * 512;       // ROWS*512
    float* norms       = slopes_proj + (size_t)ROWS * 512;    // RH
    float* kvs         = norms + RH;                          // 32*256*64
    float* part        = slopes_proj;  // reuse: slopes dead after fourier+norms
    float* av          = pos_proj;     // reuse: pos_proj dead after fourier

    dim3 blk(256);
    dim3 gGemm(ROWS / 32, 512 / 64);

    // 1-3: input projections
    gemm_rk512<<<gGemm, blk, 0, stream>>>(src, Wv, value);
    gemm_rk512<<<gGemm, blk, 0, stream>>>(pos, Wp, pos_proj);
    gemm_rk512<<<gGemm, blk, 0, stream>>>(slo, Wp, slopes_proj);

    // 4: row norms of slopes_proj
    norms_kernel<<<RH / 8, blk, 0, stream>>>(slopes_proj, norms);

    // 5: fourier features -> q', k' (transposed output layout)
    fourier_kernel<<<RH / 16, blk, 0, stream>>>(pos_proj, slopes_proj,
                                                scale, offs, proj,
                                                qprime, kprime);

    // 6-7: kvs = k'^T @ V (deterministic split-K)
    kvs_partial_kernel<<<dim3(32, KV_SPLIT), blk, 0, stream>>>(kprime, value, part);
    kvs_reduce_kernel<<<(32 * 256 * 64) / (256 * 4), blk, 0, stream>>>(
        (const float4*)part, (float4*)kvs);

    // 8: av = norms * (q' @ kvs)
    av_kernel<<<dim3(SEQ / 32, 32), blk, 0, stream>>>(qprime, kvs, norms, av);

    // 9: out = av @ Wo
    gemm_rk512<<<gGemm, blk, 0, stream>>>(av, Wo, out);
}